// Model_12429635355240
// MI455X (gfx1250) — compile-verified
//
#include <hip/hip_runtime.h>
#include <hip/hip_bf16.h>

// Problem constants (from reference)
#define B_DIM 8192
#define D_DIM 1024
#define L_DIM 32768
#define K_TOP 32

typedef float v2f __attribute__((ext_vector_type(2)));
typedef float v4f __attribute__((ext_vector_type(4)));
typedef float v8f __attribute__((ext_vector_type(8)));

// 32-bit LDS offset of a generic pointer that points into LDS
__device__ __forceinline__ unsigned lds_off(const void* p) {
    return (unsigned)(uintptr_t)(__attribute__((address_space(3))) const void*)p;
}

__device__ __forceinline__ void async_b128(unsigned lds, const void* g) {
    asm volatile("global_load_async_to_lds_b128 %0, %1, off"
                 :: "v"(lds), "v"((uint64_t)(uintptr_t)g) : "memory");
}

// ---------------------------------------------------------------------------
// Kernel 0: xbar = x - b_d   (pre-centering)
// ---------------------------------------------------------------------------
__global__ void center_kernel(const float* __restrict__ x,
                              const float* __restrict__ bd,
                              float* __restrict__ xbar) {
    size_t e = ((size_t)blockIdx.x * 256 + threadIdx.x) * 4;
    v4f xv = *(const v4f*)&x[e];
    v4f bv = *(const v4f*)&bd[e & (size_t)(D_DIM - 1)];
    *(v4f*)&xbar[e] = xv - bv;
}

// ---------------------------------------------------------------------------
// Kernel 1: W_dT[l][d] = W_d[d][l]   (32x32 LDS tile transpose)
// ---------------------------------------------------------------------------
__global__ void transpose_wd_kernel(const float* __restrict__ Wd,
                                    float* __restrict__ WdT) {
    __shared__ float tile[32][33];
    const int l0 = blockIdx.x * 32;
    const int d0 = blockIdx.y * 32;
    const int tx = threadIdx.x & 31;
    const int ty = threadIdx.x >> 5;           // 0..7
#pragma unroll
    for (int j = 0; j < 4; ++j) {
        int r = ty + 8 * j;
        tile[r][tx] = Wd[(size_t)(d0 + r) * L_DIM + (l0 + tx)];
    }
    __syncthreads();
#pragma unroll
    for (int j = 0; j < 4; ++j) {
        int r = ty + 8 * j;
        WdT[(size_t)(l0 + r) * D_DIM + (d0 + tx)] = tile[tx][r];
    }
}

// ---------------------------------------------------------------------------
// Kernel 2: encoder GEMM  a[b][l] = xbar[b][:] . W_e[l][:] + b_e[l]
//   WG tile: M=128 rows x N=128 latents, 256 threads = 8 waves (4 mt x 2 ng).
//   Wave owns two 16-row strips (mt*16, mt*16+64) x 4 adjacent 16-col tiles
//   -> 8 f32 accumulators; strips share B operands.  BOTH A and B tiles are
//   staged to LDS with gfx1250 async-to-LDS DMA, double-buffered so copies
//   for chunk k+1 overlap WMMA on chunk k.  Hot loop = ds_load + v_wmma only.
// ---------------------------------------------------------------------------
#define KT 32
#define APAD (KT + 4)                          // 36: conflict-free, 16B-aligned rows
__global__ void enc_gemm_kernel(const float* __restrict__ xbar,
                                const float* __restrict__ We,
                                const float* __restrict__ be,
                                float* __restrict__ a_out) {
    __shared__ float Abuf[2][128][APAD];       // 2 x 18 KB
    __shared__ float Bbuf[2][128][APAD];       // 2 x 18 KB   (Bbuf[col][k])
    const int l0 = blockIdx.x * 128;
    const int m0 = blockIdx.y * 128;
    const int tid  = threadIdx.x;
    const int wave = tid >> 5;
    const int lane = tid & 31;
    const int mt = wave >> 1;                  // 0..3
    const int ng = wave & 1;                   // 0..1
    const int n16   = lane & 15;               // N (or M) index within tile
    const int kbase = (lane >> 4) << 1;        // 0 or 2 : K sub-offset per ISA layout

    v8f c[8];
#pragma unroll
    for (int j = 0; j < 8; ++j) c[j] = (v8f){0.f,0.f,0.f,0.f,0.f,0.f,0.f,0.f};

    const int arow0 = mt * 16 + n16;           // LDS rows for A operands
    const int arow1 = arow0 + 64;
    const int bcol0 = ng * 64 + n16;           // LDS row (=column) for B operands

    // this thread's async-stage slots: 4 b128 per matrix per chunk
    unsigned ldsA[2][4], ldsB[2][4];
    const float* gA[4];
    const float* gB[4];
#pragma unroll
    for (int i = 0; i < 4; ++i) {
        int idx = tid + i * 256;               // 0..1023
        int row = idx >> 3;                    // KT/4 = 8 float4 per row
        int c4  = (idx & 7) << 2;
        ldsA[0][i] = lds_off(&Abuf[0][row][c4]);
        ldsA[1][i] = lds_off(&Abuf[1][row][c4]);
        ldsB[0][i] = lds_off(&Bbuf[0][row][c4]);
        ldsB[1][i] = lds_off(&Bbuf[1][row][c4]);
        gA[i] = &xbar[(size_t)(m0 + row) * D_DIM + c4];
        gB[i] = &We  [(size_t)(l0 + row) * D_DIM + c4];
    }

    // prologue: stage chunk 0 into buffer 0
#pragma unroll
    for (int i = 0; i < 4; ++i) {
        async_b128(ldsA[0][i], gA[i]);
        async_b128(ldsB[0][i], gB[i]);
    }
    asm volatile("s_wait_asynccnt 0x0" ::: "memory");
    __syncthreads();

    int cur = 0;
    for (int k0 = 0; k0 < D_DIM; k0 += KT) {
        // kick off next chunk's DMA into the other buffer (overlaps WMMA)
        if (k0 + KT < D_DIM) {
            int nxt = cur ^ 1;
#pragma unroll
            for (int i = 0; i < 4; ++i) {
                async_b128(ldsA[nxt][i], gA[i] + (k0 + KT));
                async_b128(ldsB[nxt][i], gB[i] + (k0 + KT));
            }
        }

        const float (*At)[APAD] = Abuf[cur];
        const float (*Bt)[APAD] = Bbuf[cur];
#pragma unroll
        for (int kk = 0; kk < KT; kk += 4) {
            v2f a0 = *(const v2f*)&At[arow0][kk + kbase];
            v2f a1 = *(const v2f*)&At[arow1][kk + kbase];
#pragma unroll
            for (int j = 0; j < 4; ++j) {
                v2f b = *(const v2f*)&Bt[bcol0 + j * 16][kk + kbase];
                c[j] = __builtin_amdgcn_wmma_f32_16x16x4_f32(
                    false, a0, false, b, (short)0, c[j], false, false);
                c[j + 4] = __builtin_amdgcn_wmma_f32_16x16x4_f32(
                    false, a1, false, b, (short)0, c[j + 4], false, false);
            }
        }

        // next buffer fully written (own DMAs) + everyone done reading cur
        asm volatile("s_wait_asynccnt 0x0" ::: "memory");
        __syncthreads();
        cur ^= 1;
    }

    // bias + store. C layout: VGPR r -> M = r + 8*(lane>=16), N = lane&15
    const int rb0 = m0 + mt * 16 + ((lane >> 4) << 3);
#pragma unroll
    for (int j = 0; j < 4; ++j) {
        int col = l0 + ng * 64 + j * 16 + n16;
        float bias = be[col];
#pragma unroll
        for (int r = 0; r < 8; ++r) {
            a_out[(size_t)(rb0 + r) * L_DIM + col] = c[j][r] + bias;
            a_out[(size_t)(rb0 + 64 + r) * L_DIM + col] = c[j + 4][r] + bias;
        }
    }
}

// ---------------------------------------------------------------------------
// Kernel 3: exact top-K per row via 4-pass radix select over LDS-resident
//   keys (row fits in gfx1250's 320KB LDS), then deterministic index-ordered
//   selection.  Rewrites row in place: relu(val) at top-K, 0 elsewhere.
// ---------------------------------------------------------------------------
__device__ __forceinline__ unsigned f2key(float v) {
    unsigned u = __float_as_uint(v);
    return (u & 0x80000000u) ? ~u : (u | 0x80000000u);
}
__device__ __forceinline__ float key2f(unsigned k) {
    unsigned u = (k & 0x80000000u) ? (k ^ 0x80000000u) : ~k;
    return __uint_as_float(u);
}

__global__ void topk_select_kernel(float* __restrict__ a_f,
                                   float* __restrict__ vals,
                                   int* __restrict__ idxs) {
    __shared__ unsigned keys[L_DIM];           // 128 KB: whole row as sortable keys
    __shared__ unsigned hist[256];
    __shared__ unsigned sh_prefix;
    __shared__ int sh_want;
    __shared__ int cg[256], ct[256];

    const int b   = blockIdx.x;
    const int tid = threadIdx.x;
    float* row = a_f + (size_t)b * L_DIM;

    for (int i = tid; i < L_DIM; i += 256)
        keys[i] = f2key(row[i]);

    if (tid == 0) { sh_prefix = 0u; sh_want = K_TOP; }
    unsigned mask = 0u;
    __syncthreads();

    for (int shift = 24; shift >= 0; shift -= 8) {
        hist[tid] = 0u;
        __syncthreads();
        unsigned prefix = sh_prefix;
        for (int i = tid; i < L_DIM; i += 256) {
            unsigned u = keys[i];
            if ((u & mask) == prefix)
                atomicAdd(&hist[(u >> shift) & 255u], 1u);
        }
        __syncthreads();
        if (tid == 0) {
            int w = sh_want;
            unsigned p = prefix;
            for (int bin = 255; bin >= 0; --bin) {
                int cnt = (int)hist[bin];
                if (cnt >= w) { p |= ((unsigned)bin << shift); break; }
                w -= cnt;
            }
            sh_prefix = p;
            sh_want = w;
        }
        mask |= (0xFFu << shift);
        __syncthreads();
    }

    const unsigned kth = sh_prefix;            // exact K-th largest key
    const int tiesAllowed = sh_want;           // ties (== kth) to accept

    // blocked deterministic selection: thread t owns [t*128, t*128+128)
    const int CH = L_DIM / 256;                // 128
    const int base = tid * CH;
    int g = 0, t = 0;
    for (int i = 0; i < CH; ++i) {
        unsigned u = keys[base + i];
        g += (u > kth);
        t += (u == kth);
    }
    cg[tid] = g; ct[tid] = t;
    __syncthreads();
    if (tid == 0) {                            // exclusive scans (256 elems, cheap)
        int ag = 0, at = 0;
        for (int i = 0; i < 256; ++i) {
            int tg = cg[i], tt = ct[i];
            cg[i] = ag; ct[i] = at;
            ag += tg; at += tt;
        }
    }
    __syncthreads();
    int slotG   = cg[tid];                     // #greater at smaller index
    int tieRank = ct[tid];                     // #ties    at smaller index

    for (int i = 0; i < CH; ++i) {
        unsigned u = keys[base + i];
        float fo = 0.f;
        if (u > kth) {
            int accTies = tieRank < tiesAllowed ? tieRank : tiesAllowed;
            int slot = slotG + accTies;
            fo = fmaxf(key2f(u), 0.f);
            vals[(size_t)b * K_TOP + slot] = fo;
            idxs[(size_t)b * K_TOP + slot] = base + i;
            ++slotG;
        } else if (u == kth) {
            if (tieRank < tiesAllowed) {
                int slot = slotG + tieRank;
                fo = fmaxf(key2f(u), 0.f);
                vals[(size_t)b * K_TOP + slot] = fo;
                idxs[(size_t)b * K_TOP + slot] = base + i;
            }
            ++tieRank;
        }
        row[base + i] = fo;
    }
}

// ---------------------------------------------------------------------------
// Kernel 4: sparse decoder  xhat[b][:] = b_d + sum_k val_k * W_dT[idx_k][:]
// ---------------------------------------------------------------------------
__global__ void decode_kernel(const float* __restrict__ vals,
                              const int* __restrict__ idxs,
                              const float* __restrict__ WdT,
                              const float* __restrict__ bd,
                              float* __restrict__ xhat) {
    __shared__ float sv[K_TOP];
    __shared__ int   si[K_TOP];
    const int b   = blockIdx.x;
    const int tid = threadIdx.x;
    if (tid < K_TOP) {
        sv[tid] = vals[(size_t)b * K_TOP + tid];
        si[tid] = idxs[(size_t)b * K_TOP + tid];
    }
    __syncthreads();
    const int d = tid * 4;                     // 256 threads * 4 = 1024 = D
    v4f acc = *(const v4f*)&bd[d];
#pragma unroll 4
    for (int k = 0; k < K_TOP; ++k) {
        float v = sv[k];
        v4f w = *(const v4f*)&WdT[(size_t)si[k] * D_DIM + d];
        acc = acc + w * v;
    }
    *(v4f*)&xhat[(size_t)b * D_DIM + d] = acc;
}

// ---------------------------------------------------------------------------
// Host launcher
// ---------------------------------------------------------------------------
extern "C" void kernel_launch(void* const* d_in, const int* in_sizes, int n_in,
                              void* d_out, int out_size, void* d_ws, size_t ws_size,
                              hipStream_t stream) {
    (void)in_sizes; (void)n_in; (void)out_size; (void)ws_size;

    const float* x  = (const float*)d_in[0];
    const float* We = (const float*)d_in[1];
    const float* be = (const float*)d_in[2];
    const float* Wd = (const float*)d_in[3];
    const float* bd = (const float*)d_in[4];
    // d_in[5] = K scalar; reference fixes K = 32 (K_TOP)

    float* xhat = (float*)d_out;
    float* f    = (float*)d_out + (size_t)B_DIM * D_DIM;   // dense a, then f, in place

    const size_t XBAR_BYTES = (size_t)B_DIM * D_DIM * 4;     // 32 MB
    const size_t VAL_BYTES  = (size_t)B_DIM * K_TOP * 4;     // 1 MB
    const size_t IDX_BYTES  = (size_t)B_DIM * K_TOP * 4;     // 1 MB
    char* ws = (char*)d_ws;
    float* xbar = (float*)ws;
    float* vals = (float*)(ws + XBAR_BYTES);
    int*   idxs = (int*)  (ws + XBAR_BYTES + VAL_BYTES);
    float* WdT  = (float*)(ws + XBAR_BYTES + VAL_BYTES + IDX_BYTES);  // 128 MB

    // 0) xbar = x - b_d
    center_kernel<<<(B_DIM * D_DIM) / (256 * 4), 256, 0, stream>>>(x, bd, xbar);

    // 1) transpose W_d -> W_dT
    transpose_wd_kernel<<<dim3(L_DIM / 32, D_DIM / 32), 256, 0, stream>>>(Wd, WdT);

    // 2) encoder GEMM (WMMA f32), dense pre-activations into f region
    enc_gemm_kernel<<<dim3(L_DIM / 128, B_DIM / 128), 256, 0, stream>>>(xbar, We, be, f);

    // 3) exact top-K + relu, in-place sparsification, compact list to ws
    topk_select_kernel<<<B_DIM, 256, 0, stream>>>(f, vals, idxs);

    // 4) sparse decoder
    decode_kernel<<<B_DIM, 256, 0, stream>>>(vals, idxs, WdT, bd, xhat);
}